// DeepSeekV3MoEGate_45947560133085
// MI455X (gfx1250) — compile-verified
//
#include <hip/hip_runtime.h>
#include <math.h>

// ---------------- problem constants ----------------
#define T_TOKENS  8192      // 4 * 2048
#define H_DIM     7168
#define NEXP      256
#define TOPK      8
#define NGROUP    8         // 32 experts per group
#define TOPKGRP   4
#define SCALE     2.5f

typedef float v2f __attribute__((ext_vector_type(2)));
typedef float v8f __attribute__((ext_vector_type(8)));
typedef unsigned int v4u __attribute__((ext_vector_type(4)));
typedef int          v4i __attribute__((ext_vector_type(4)));
typedef int          v8i __attribute__((ext_vector_type(8)));

// ---------------- GEMM: logits[t][e] = sum_h x[t][h] * w[e][h] ----------------
// Block tile: M=64 rows x N=256 experts, K staged 32 at a time through LDS via
// the Tensor Data Mover (double-buffered, TENSORcnt-tracked).
#define BM 64
#define BK 32
#define LPITCH (BK + 4)     // 36 floats: TDM pad -> conflict-free LDS rows

// Issue one TDM 2D-tile load: nrows x BK floats, row stride H_DIM floats,
// into LDS at lds_off with 4-DWORD padding after every 32-DWORD row.
__device__ __forceinline__ void tdm_load_tile(const float* gbase,
                                              unsigned lds_off, int nrows) {
  unsigned long long ga = (unsigned long long)(uintptr_t)gbase;
  v4u g0;
  g0.x = 1u;                                   // count=1, user desc, no gather
  g0.y = lds_off;                              // LDS byte address
  g0.z = (unsigned)ga;                         // global_addr[31:0]
  g0.w = ((unsigned)(ga >> 32) & 0x01FFFFFFu)  // global_addr[56:32]
         | (2u << 30);                         // type = 2 ("image")
  v8i g1;
  g1[0] = (2 << 16)      // data_size = 4 bytes
        | (1 << 20)      // pad_enable
        | (4 << 22)      // pad_interval: 2^(4+1) = 32 DWORDs
        | (3 << 25);     // pad_amount: 3+1 = 4 DWORDs
  g1[1] = (BK << 16);            // tensor_dim0[15:0] = 32
  g1[2] = (nrows << 16);         // tensor_dim0 hi = 0 | tensor_dim1[15:0]
  g1[3] = (BK << 16);            // tensor_dim1 hi = 0 | tile_dim0 = 32
  g1[4] = nrows;                 // tile_dim1 = rows, tile_dim2 = 0
  g1[5] = H_DIM;                 // tensor_dim0_stride[31:0]
  g1[6] = 0;                     // stride0 hi, stride1 lo
  g1[7] = 0;
  v4i gz4 = {0, 0, 0, 0};
  v8i gz8 = {0, 0, 0, 0, 0, 0, 0, 0};
  __builtin_amdgcn_tensor_load_to_lds(g0, g1, gz4, gz4, gz8, 0);
}

__global__ __launch_bounds__(256)
void moe_gate_gemm(const float* __restrict__ x,
                   const float* __restrict__ w,
                   float* __restrict__ logits) {
  __shared__ float Xs[2][BM][LPITCH];      // 2 *  9216 B
  __shared__ float Ws[2][NEXP][LPITCH];    // 2 * 36864 B  (total ~92 KB)

  const int tid  = threadIdx.x;
  const int lane = tid & 31;
  const int wv   = tid >> 5;
  const int rowBase = blockIdx.x * BM;
  const int nBase   = wv * 32;

  v8f acc[4][2];
  #pragma unroll
  for (int i = 0; i < 4; ++i)
    #pragma unroll
    for (int j = 0; j < 2; ++j)
      acc[i][j] = v8f{};

  const int mrow = lane & 15;          // A: M row / B: N col within 16-tile
  const int koff = (lane >> 4) * 2;    // lanes 16-31 carry K+2,K+3

  const int NITER = H_DIM / BK;        // 224

  // prologue: DMA slab 0 into buffer 0 (wave 0 only)
  if (wv == 0) {
    tdm_load_tile(x + (size_t)rowBase * H_DIM, (unsigned)(uintptr_t)&Xs[0][0][0], BM);
    tdm_load_tile(w,                           (unsigned)(uintptr_t)&Ws[0][0][0], NEXP);
  }

  for (int it = 0; it < NITER; ++it) {
    const int buf = it & 1;
    if (wv == 0) {
      if (it + 1 < NITER) {
        const int k1 = (it + 1) * BK;
        tdm_load_tile(x + (size_t)rowBase * H_DIM + k1,
                      (unsigned)(uintptr_t)&Xs[buf ^ 1][0][0], BM);
        tdm_load_tile(w + k1,
                      (unsigned)(uintptr_t)&Ws[buf ^ 1][0][0], NEXP);
        __builtin_amdgcn_s_wait_tensorcnt(2);  // current slab's pair done
      } else {
        __builtin_amdgcn_s_wait_tensorcnt(0);
      }
    }
    __syncthreads();                           // slab `buf` ready for all waves

    #pragma unroll
    for (int k = 0; k < BK; k += 4) {
      v2f a[4], b[2];
      #pragma unroll
      for (int mt = 0; mt < 4; ++mt)
        a[mt] = *reinterpret_cast<const v2f*>(&Xs[buf][mt * 16 + mrow][k + koff]);
      #pragma unroll
      for (int nt = 0; nt < 2; ++nt)
        b[nt] = *reinterpret_cast<const v2f*>(&Ws[buf][nBase + nt * 16 + mrow][k + koff]);
      #pragma unroll
      for (int mt = 0; mt < 4; ++mt)
        #pragma unroll
        for (int nt = 0; nt < 2; ++nt)
          acc[mt][nt] = __builtin_amdgcn_wmma_f32_16x16x4_f32(
              false, a[mt], false, b[nt], (short)0, acc[mt][nt], false, false);
    }
    __syncthreads();                           // all reads of `buf` done
  }

  // ---- store: C/D layout: VGPR r -> M = 16*mt + r + 8*(lane>=16), N = lane&15 ----
  const int colIn = lane & 15;
  const int mHalf = (lane >> 4) * 8;
  #pragma unroll
  for (int mt = 0; mt < 4; ++mt)
    #pragma unroll
    for (int nt = 0; nt < 2; ++nt) {
      int col = nBase + nt * 16 + colIn;
      #pragma unroll
      for (int r = 0; r < 8; ++r) {
        int row = rowBase + mt * 16 + mHalf + r;
        logits[(size_t)row * NEXP + col] = acc[mt][nt][r];
      }
    }
}

// ---------------- top-k gate: one wave32 per token ----------------
__global__ __launch_bounds__(256)
void moe_gate_topk(const float* __restrict__ logits,
                   const float* __restrict__ bias,
                   int* __restrict__ out_idx,
                   float* __restrict__ out_w) {
  const int lane = threadIdx.x & 31;
  const int wv   = threadIdx.x >> 5;
  const int t    = blockIdx.x * 8 + wv;

  const float* row = logits + (size_t)t * NEXP;

  // lane owns experts [8*lane, 8*lane+8); group g = lane quad (lane/4)
  float s[8], sb[8];
  #pragma unroll
  for (int i = 0; i < 8; ++i) {
    float lg = row[lane * 8 + i];
    float sv = 1.0f / (1.0f + __expf(-lg));
    s[i]  = sv;
    sb[i] = sv + bias[lane * 8 + i];
  }

  // lane-local top-2 of bias-corrected scores
  float m1 = -INFINITY, m2 = -INFINITY;
  #pragma unroll
  for (int i = 0; i < 8; ++i) {
    float v = sb[i];
    if (v > m1) { m2 = m1; m1 = v; }
    else if (v > m2) { m2 = v; }
  }
  // merge top-2 across the 4 lanes of the group
  #pragma unroll
  for (int off = 1; off <= 2; off <<= 1) {
    float o1 = __shfl_xor(m1, off, 32);
    float o2 = __shfl_xor(m2, off, 32);
    float lo  = fminf(m1, o1);
    float sec = (m1 >= o1) ? m2 : o2;
    m1 = fmaxf(m1, o1);
    m2 = fmaxf(lo, sec);
  }
  float gs = m1 + m2;                       // group score (valid on all quad lanes)

  // group scores to lanes 0..7; rank-select top-4 groups (ties -> lower index)
  float gsl = __shfl(gs, (lane < 8 ? lane * 4 : 0), 32);
  int rank = 0;
  #pragma unroll
  for (int j = 0; j < NGROUP; ++j) {
    float gj = __shfl(gs, j * 4, 32);
    if (lane < 8)
      rank += (gj > gsl) || (gj == gsl && j < lane);
  }
  bool sel = (lane < 8) && (rank < TOPKGRP);
  unsigned gmask = (unsigned)__ballot(sel) & 0xFFu;

  const bool myGroupSel = (gmask >> (lane >> 2)) & 1u;
  float mval[8];
  #pragma unroll
  for (int i = 0; i < 8; ++i)
    mval[i] = myGroupSel ? sb[i] : -INFINITY;

  // 8 rounds of wave-wide argmax (lower index wins ties)
  float tw[TOPK]; int tidx[TOPK]; float wsum = 0.0f;
  #pragma unroll
  for (int kk = 0; kk < TOPK; ++kk) {
    float bv = mval[0]; int bi = 0;
    #pragma unroll
    for (int i = 1; i < 8; ++i)
      if (mval[i] > bv) { bv = mval[i]; bi = i; }
    int bidx = lane * 8 + bi;
    #pragma unroll
    for (int off = 16; off >= 1; off >>= 1) {
      float ov = __shfl_xor(bv, off, 32);
      int   oi = __shfl_xor(bidx, off, 32);
      if (ov > bv || (ov == bv && oi < bidx)) { bv = ov; bidx = oi; }
    }
    // un-biased sigmoid score of winner via shuffle from owning lane
    int owner = bidx >> 3, slot = bidx & 7;
    float sv = s[0];
    #pragma unroll
    for (int i = 1; i < 8; ++i) sv = (slot == i) ? s[i] : sv;
    sv = __shfl(sv, owner, 32);
    tw[kk] = sv; tidx[kk] = bidx; wsum += sv;
    if (lane == owner) {
      #pragma unroll
      for (int i = 0; i < 8; ++i)
        if (slot == i) mval[i] = -INFINITY;
    }
  }

  const float inv = SCALE / wsum;
  if (lane == 0) {
    #pragma unroll
    for (int kk = 0; kk < TOPK; ++kk) {
      out_idx[(size_t)t * TOPK + kk] = tidx[kk];
      out_w[(size_t)t * TOPK + kk]   = tw[kk] * inv;
    }
  }
}

// ---------------- launch ----------------
extern "C" void kernel_launch(void* const* d_in, const int* in_sizes, int n_in,
                              void* d_out, int out_size, void* d_ws, size_t ws_size,
                              hipStream_t stream) {
  const float* x    = (const float*)d_in[0];   // [4,2048,7168] fp32
  const float* w    = (const float*)d_in[1];   // [256,7168]    fp32
  const float* bias = (const float*)d_in[2];   // [256]         fp32

  float* logits = (float*)d_ws;                          // 8192*256*4 = 8 MB
  int*   out_idx = (int*)d_out;                          // first T*8 elems (int32)
  float* out_w   = (float*)d_out + (size_t)T_TOKENS * TOPK;

  moe_gate_gemm<<<T_TOKENS / BM, 256, 0, stream>>>(x, w, logits);
  moe_gate_topk<<<T_TOKENS / 8, 256, 0, stream>>>(logits, bias, out_idx, out_w);
}